// CRFLayer_18184891531946
// MI455X (gfx1250) — compile-verified
//
#include <hip/hip_runtime.h>
#include <hip/hip_bf16.h>
#include <math.h>

// Problem sizes (match reference)
#define Bsz 256
#define Ssz 2048
#define Dsz 64
#define TB  16        // batches per workgroup (one 16-row WMMA tile)
#define PSTR 68       // padded LDS row stride (floats) -> conflict-free banks
#define RNMASK 3      // renormalize when (t & RNMASK) == 0  (every 4 steps)
#define LOG2E 1.44269504088896340736f
#define LN2   0.69314718055994530942f

typedef __attribute__((ext_vector_type(2))) float v2f;
typedef __attribute__((ext_vector_type(8))) float v8f;

// ---------------------------------------------------------------------------
// Kernel 1: target[b] = point_score + trans_score  (gather + block reduction)
// ---------------------------------------------------------------------------
__global__ __launch_bounds__(256) void crf_target_kernel(
    const float* __restrict__ yp, const int* __restrict__ yt,
    const float* __restrict__ mask, const float* __restrict__ trans,
    float* __restrict__ tgt)
{
  const int b = blockIdx.x;
  const int tid = threadIdx.x;
  float acc = 0.0f;
  for (int n = tid; n < Ssz; n += blockDim.x) {
    float m  = mask[(size_t)b * Ssz + n];
    int   l  = yt[(size_t)b * Ssz + n];
    acc += m * m * yp[((size_t)b * Ssz + n) * Dsz + l];
    if (n + 1 < Ssz) {
      float m2 = mask[(size_t)b * Ssz + n + 1];
      int   l2 = yt[(size_t)b * Ssz + n + 1];
      acc += m * m2 * trans[l * Dsz + l2];
    }
  }
  __shared__ float red[256];
  red[tid] = acc;
  __syncthreads();
  for (int s = 128; s > 0; s >>= 1) {
    if (tid < s) red[tid] += red[tid + s];
    __syncthreads();
  }
  if (tid == 0) tgt[b] = red[0];
}

// ---------------------------------------------------------------------------
// Kernel 2: forward scan, base-2 log domain.  4 waves / block, 16 batches /
// block, grid = B/16.  Per step: 16 chained v_wmma_f32_16x16x4_f32 (K=64),
// E=exp(trans) in registers.  Raw v_log_f32 / v_exp_f32 (log2/exp2), state
// renormalized every 4 steps, Pbuf double-buffered -> 1 barrier per step.
// ---------------------------------------------------------------------------
__global__ __launch_bounds__(128) void crf_scan_kernel(
    const float* __restrict__ yp, const float* __restrict__ mask,
    const float* __restrict__ trans, const float* __restrict__ tgt,
    float* __restrict__ out)
{
  __shared__ float Pbuf[2][TB * PSTR]; // exp2(state), double-buffered
  __shared__ float wmax[4][TB];        // per-wave row maxima (renorm steps)
  __shared__ float rowM[TB];           // combined per-row max
  __shared__ float rowOff[TB];         // accumulated log2-offset per batch

  const int tid  = threadIdx.x;
  const int wave = tid >> 5;
  const int lane = tid & 31;
  const int half = lane >> 4;   // 0: rows 0..7 / K 4k+0,1   1: rows 8..15 / K 4k+2,3
  const int ln   = lane & 15;
  const int cw   = wave * 16;   // this wave's state-column base
  const int col  = cw + ln;     // this lane's column in C layout
  const int b0   = blockIdx.x * TB;

  // B fragments: E[k][col] = exp(trans) = exp2(trans * log2e), constant.
  v2f bfrag[16];
  #pragma unroll
  for (int kk = 0; kk < 16; ++kk) {
    int k0 = 4 * kk + 2 * half;
    bfrag[kk].x = __builtin_amdgcn_exp2f(trans[(k0    ) * Dsz + col] * LOG2E);
    bfrag[kk].y = __builtin_amdgcn_exp2f(trans[(k0 + 1) * Dsz + col] * LOG2E);
  }

  if (tid < TB) rowOff[tid] = 0.0f;
  __syncthreads();

  float s[8];   // state in log2 units, C layout: row = r + 8*half, this col

  // ---- t = 0: state0 = yp[:,0]*m0 (log2 units), renormalize, seed Pbuf[0]
  {
    float sn[8];
    #pragma unroll
    for (int r = 0; r < 8; ++r) {
      int row = r + 8 * half;
      float cm = mask[(size_t)(b0 + row) * Ssz + 0];
      sn[r] = yp[((size_t)(b0 + row) * Ssz + 0) * Dsz + col] * (cm * LOG2E);
    }
    float m8[8];
    #pragma unroll
    for (int r = 0; r < 8; ++r) m8[r] = sn[r];
    #pragma unroll
    for (int d = 1; d < 16; d <<= 1) {
      #pragma unroll
      for (int r = 0; r < 8; ++r) m8[r] = fmaxf(m8[r], __shfl_xor(m8[r], d, 32));
    }
    if (ln == 0) {
      #pragma unroll
      for (int r = 0; r < 8; ++r) wmax[wave][r + 8 * half] = m8[r];
    }
    __syncthreads();
    if (tid < TB) {
      float rm = fmaxf(fmaxf(wmax[0][tid], wmax[1][tid]),
                       fmaxf(wmax[2][tid], wmax[3][tid]));
      rowM[tid] = rm;
      rowOff[tid] += rm;
    }
    __syncthreads();
    #pragma unroll
    for (int r = 0; r < 8; ++r) {
      int row = r + 8 * half;
      s[r] = sn[r] - rowM[row];
      Pbuf[0][row * PSTR + col] = __builtin_amdgcn_exp2f(s[r]);
    }
    __syncthreads();
  }

  // ---- t = 1 .. S-1  (read Pbuf[(t-1)&1], write Pbuf[t&1]; 1 barrier/step)
  for (int t = 1; t < Ssz; ++t) {
    const int rd = (t - 1) & 1;
    const int wr = t & 1;

    // Prefetch next step's emission rows (hide HBM/L2 latency on the chain).
    if (ln == 0 && t + 1 < Ssz) {
      #pragma unroll
      for (int r = 0; r < 8; ++r) {
        int row = r + 8 * half;
        __builtin_prefetch(&yp[((size_t)(b0 + row) * Ssz + (t + 1)) * Dsz + cw], 0, 1);
      }
    }

    // acc[m][n] = sum_k exp2(state)[m][k] * E[k][n]   (K=64 as 16 K4 WMMAs)
    v8f acc = {};
    #pragma unroll
    for (int kk = 0; kk < 16; ++kk) {
      int k0 = 4 * kk + 2 * half;
      v2f a = *(const v2f*)&Pbuf[rd][ln * PSTR + k0];
      acc = __builtin_amdgcn_wmma_f32_16x16x4_f32(
                false, a, false, bfrag[kk], (short)0, acc, false, false);
    }

    // w = log2(acc) + e*cm*log2e ; masked update (shift-invariant in log2)
    float sn[8];
    #pragma unroll
    for (int r = 0; r < 8; ++r) {
      int row = r + 8 * half;
      float cm = mask[(size_t)(b0 + row) * Ssz + t];
      float e  = yp[((size_t)(b0 + row) * Ssz + t) * Dsz + col] * (cm * LOG2E);
      float w  = __builtin_amdgcn_logf(acc[r]) + e;   // raw v_log_f32 (log2)
      sn[r] = cm * w + (1.0f - cm) * s[r];
    }

    if ((t & RNMASK) == 0) {
      // Renormalize: per-batch max over 64 states (butterfly + LDS combine).
      float m8[8];
      #pragma unroll
      for (int r = 0; r < 8; ++r) m8[r] = sn[r];
      #pragma unroll
      for (int d = 1; d < 16; d <<= 1) {
        #pragma unroll
        for (int r = 0; r < 8; ++r) m8[r] = fmaxf(m8[r], __shfl_xor(m8[r], d, 32));
      }
      if (ln == 0) {
        #pragma unroll
        for (int r = 0; r < 8; ++r) wmax[wave][r + 8 * half] = m8[r];
      }
      __syncthreads();
      if (tid < TB) {
        float rm = fmaxf(fmaxf(wmax[0][tid], wmax[1][tid]),
                         fmaxf(wmax[2][tid], wmax[3][tid]));
        rowM[tid] = rm;
        rowOff[tid] += rm;
      }
      __syncthreads();
      #pragma unroll
      for (int r = 0; r < 8; ++r) {
        int row = r + 8 * half;
        float v = sn[r] - rowM[row];
        s[r] = v;
        Pbuf[wr][row * PSTR + col] = __builtin_amdgcn_exp2f(v);
      }
    } else {
      #pragma unroll
      for (int r = 0; r < 8; ++r) {
        s[r] = sn[r];
        Pbuf[wr][(r + 8 * half) * PSTR + col] = __builtin_amdgcn_exp2f(sn[r]);
      }
    }
    __syncthreads();   // make Pbuf[wr] visible for step t+1
  }

  // ---- finalize: out[b] = ln2*(offset + log2(sum_j exp2(s_j))) - target[b]
  if (tid < TB) {
    const int last = (Ssz - 1) & 1;
    float sum = 0.0f;
    #pragma unroll
    for (int j = 0; j < Dsz; ++j) sum += Pbuf[last][tid * PSTR + j];
    out[b0 + tid] = (rowOff[tid] + __builtin_amdgcn_logf(sum)) * LN2 - tgt[b0 + tid];
  }
}

// ---------------------------------------------------------------------------
extern "C" void kernel_launch(void* const* d_in, const int* in_sizes, int n_in,
                              void* d_out, int out_size, void* d_ws, size_t ws_size,
                              hipStream_t stream) {
  const float* y_pred = (const float*)d_in[0];
  const int*   y_true = (const int*)  d_in[1];
  const float* mask   = (const float*)d_in[2];
  const float* trans  = (const float*)d_in[3];
  float* out = (float*)d_out;
  float* tgt = (float*)d_ws;            // B floats of scratch

  crf_target_kernel<<<Bsz, 256, 0, stream>>>(y_pred, y_true, mask, trans, tgt);
  crf_scan_kernel<<<Bsz / TB, 128, 0, stream>>>(y_pred, mask, trans, tgt, out);
}